// Top2Router_10642928959654
// MI455X (gfx1250) — compile-verified
//
#include <hip/hip_runtime.h>
#include <hip/hip_bf16.h>

// Problem constants from the reference: B=4, S=4096, H=2048, E=16
#define RT_H 2048
#define RT_E 16
#define RT_NTOK 16384              // B*S
#define WAVES_PER_BLOCK 8
#define TILES (RT_NTOK / 16)       // 1024 tiles of 16 tokens
#define KPANEL 512                 // W staged in 4 panels of 512 columns
#define PW (KPANEL + 4)            // padded LDS row stride (words): 516 % 64 == 4
                                   //   -> lane lm reads banks 4*lm..4*lm+3: conflict-free

typedef __attribute__((ext_vector_type(2))) float v2f;
typedef __attribute__((ext_vector_type(4))) float v4f;
typedef __attribute__((ext_vector_type(8))) float v8f;

__global__ __launch_bounds__(256) void top2router_kernel(
    const float* __restrict__ x,   // [16384, 2048]
    const float* __restrict__ W,   // [16, 2048]
    const float* __restrict__ b,   // [16]
    float* __restrict__ out)       // [16384*2] weights then [16384*2] indices(as float)
{
    const int tid  = threadIdx.x;
    const int lane = tid & 31;
    const int wave = tid >> 5;
    const int half = lane >> 4;        // 0: K=k0..k0+1, 1: K=k0+2..k0+3 within a 4-chunk
    const int lm   = lane & 15;        // A: row M; B: column N (expert)
    const int tile = blockIdx.x * WAVES_PER_BLOCK + wave;
    const int m0   = tile * 16;

    __shared__ float s_w[RT_E * PW];            // 33 KB: one 16 x 512 panel of W (padded)
    __shared__ float s_logits[WAVES_PER_BLOCK * 256]; // 8 KB

    const float bias = b[lm];

    // A-fragment addressing (ISA 7.12.2, 32-bit A 16x4):
    //   lane<16 -> (M=lm, K=k0,k0+1) ; lane>=16 -> (M=lm, K=k0+2,k0+3)
    const float* xrow = x + (size_t)(m0 + lm) * RT_H + (half ? 2 : 0);
    // B-fragment comes from LDS: W[lm][k0 + 2*half .. +1]
    const int wbase = lm * PW + 2 * half;

    v8f acc0, acc1, acc2, acc3;
#pragma unroll
    for (int r = 0; r < 8; ++r) {
        acc0[r] = bias;   // fold bias into one accumulator
        acc1[r] = 0.0f;
        acc2[r] = 0.0f;
        acc3[r] = 0.0f;
    }

    for (int p = 0; p < RT_H / KPANEL; ++p) {
        const int kbase = p * KPANEL;

        // ---- Stage W[:, kbase:kbase+512] into LDS (coalesced b128 -> ds b128) ----
        if (p != 0) __syncthreads();   // previous panel fully consumed
#pragma unroll
        for (int i = 0; i < 8; ++i) {
            const int f4  = i * 256 + tid;    // 0..2047 float4s (16 rows x 128)
            const int row = f4 >> 7;
            const int c4  = f4 & 127;
            v4f val = *(const v4f*)(W + (size_t)row * RT_H + kbase + c4 * 4);
            *(v4f*)(&s_w[row * PW + c4 * 4]) = val;
        }
        __syncthreads();

        // ---- Hot loop: 1 global b64 (x) + 1 ds b64 (W) per WMMA ----
        for (int kk = 0; kk < KPANEL; kk += 16) {
            const int k = kbase + kk;
            __builtin_prefetch(xrow + k + 512, 0, 3);  // pull streamed x toward WGP

            v2f a0 = *(const v2f*)(xrow + k +  0);
            v2f w0 = *(const v2f*)(&s_w[wbase + kk +  0]);
            v2f a1 = *(const v2f*)(xrow + k +  4);
            v2f w1 = *(const v2f*)(&s_w[wbase + kk +  4]);
            v2f a2 = *(const v2f*)(xrow + k +  8);
            v2f w2 = *(const v2f*)(&s_w[wbase + kk +  8]);
            v2f a3 = *(const v2f*)(xrow + k + 12);
            v2f w3 = *(const v2f*)(&s_w[wbase + kk + 12]);

            acc0 = __builtin_amdgcn_wmma_f32_16x16x4_f32(false, a0, false, w0,
                                                         (short)0, acc0, false, false);
            acc1 = __builtin_amdgcn_wmma_f32_16x16x4_f32(false, a1, false, w1,
                                                         (short)0, acc1, false, false);
            acc2 = __builtin_amdgcn_wmma_f32_16x16x4_f32(false, a2, false, w2,
                                                         (short)0, acc2, false, false);
            acc3 = __builtin_amdgcn_wmma_f32_16x16x4_f32(false, a3, false, w3,
                                                         (short)0, acc3, false, false);
        }
    }

    v8f c;
#pragma unroll
    for (int r = 0; r < 8; ++r)
        c[r] = (acc0[r] + acc1[r]) + (acc2[r] + acc3[r]);

    // ---- Stage logits to LDS so one lane can see a whole token row ----
    // C/D layout: VGPR r holds row (half ? r+8 : r), column = lm.
    const int base = wave * 256;
#pragma unroll
    for (int r = 0; r < 8; ++r)
        s_logits[base + (half * 8 + r) * 16 + lm] = c[r];
    __syncthreads();

    // ---- Softmax + top-2 + renorm: lanes 0..15 each own one token ----
    if (lane < 16) {
        float v[RT_E];
#pragma unroll
        for (int e = 0; e < RT_E; ++e)
            v[e] = s_logits[base + lane * 16 + e];

        float m = v[0];
#pragma unroll
        for (int e = 1; e < RT_E; ++e) m = fmaxf(m, v[e]);

        float Z = 0.0f;
#pragma unroll
        for (int e = 0; e < RT_E; ++e) Z += __expf(v[e] - m);
        const float rZ = __frcp_rn(Z);

        // top-2 scan, strict > keeps the lowest index on ties (JAX top_k order)
        float v1 = -3.402823466e+38f, v2 = -3.402823466e+38f;
        int   i1 = 0, i2 = 0;
#pragma unroll
        for (int e = 0; e < RT_E; ++e) {
            float val = v[e];
            if (val > v1)      { v2 = v1; i2 = i1; v1 = val; i1 = e; }
            else if (val > v2) { v2 = val; i2 = e; }
        }

        const float p1 = __expf(v1 - m) * rZ;
        const float p2 = __expf(v2 - m) * rZ;
        const float rs = __frcp_rn(p1 + p2 + 1e-6f);

        const int token = m0 + lane;
        // output 0: top_weights [NTOK,2]
        out[token * 2 + 0] = p1 * rs;
        out[token * 2 + 1] = p2 * rs;
        // output 1: top_indices [NTOK,2] (harness views output buffer as float)
        float* oi = out + 2 * RT_NTOK;
        oi[token * 2 + 0] = (float)i1;
        oi[token * 2 + 1] = (float)i2;
    }
}

extern "C" void kernel_launch(void* const* d_in, const int* in_sizes, int n_in,
                              void* d_out, int out_size, void* d_ws, size_t ws_size,
                              hipStream_t stream) {
    (void)in_sizes; (void)n_in; (void)d_ws; (void)ws_size; (void)out_size;
    const float* x = (const float*)d_in[0];
    const float* W = (const float*)d_in[1];
    const float* b = (const float*)d_in[2];
    float* out = (float*)d_out;

    dim3 block(32 * WAVES_PER_BLOCK);            // 256 threads = 8 waves (wave32)
    dim3 grid(TILES / WAVES_PER_BLOCK);          // 1024 tiles / 8 = 128 blocks
    top2router_kernel<<<grid, block, 0, stream>>>(x, W, b, out);
}